// Style_12919261626842
// MI455X (gfx1250) — compile-verified
//
#include <hip/hip_runtime.h>

typedef float v2f __attribute__((ext_vector_type(2)));
typedef float v8f __attribute__((ext_vector_type(8)));

#define D_DIM   1024
#define NTILE   (D_DIM / 16)                   // 64 tile rows/cols
#define TRI_T   (NTILE * (NTILE + 1) / 2)      // 2080 upper-triangle tiles / sample
#define BLK_PER_SAMPLE (TRI_T / 8)             // 260 blocks (8 waves each) / sample
#define P_OUT   ((size_t)D_DIM * (D_DIM + 1) / 2)   // 524800 pairs / sample

// One wave = one 16x16 Gram tile via a single V_WMMA_F32_16X16X4_F32
// (rank-1 outer product: B's K>0 rows are zero, so only K=0 contributes).
// Each block handles 8 tiles of ONE sample so the x / out bases are
// block-uniform (SGPR) and stores take the saddr + 32-bit voffset form.
__global__ __launch_bounds__(256) void gram_triu_wmma(const float* __restrict__ x,
                                                      float* __restrict__ out) {
    const int wave = threadIdx.x >> 5;                           // 8 wave32 / block
    const int lane = threadIdx.x & 31;

    const int n = blockIdx.x / BLK_PER_SAMPLE;                   // sample (SGPR)
    const int t = (blockIdx.x - n * BLK_PER_SAMPLE) * 8 + wave;  // tile id 0..2079

    // Decode (ti, tj): rowOff(ti) = ti*NTILE - ti*(ti-1)/2 <= t < rowOff(ti+1).
    const float disc = (2.0f * NTILE + 1.0f) * (2.0f * NTILE + 1.0f) - 8.0f * (float)t;
    int ti = (int)((2.0f * NTILE + 1.0f - sqrtf(disc)) * 0.5f);
    ti = ti < 0 ? 0 : (ti > NTILE - 1 ? NTILE - 1 : ti);
    while ((ti + 1) * NTILE - ((ti + 1) * ti) / 2 <= t) ++ti;    // <=2 fixup iters
    while (ti * NTILE - (ti * (ti - 1)) / 2 > t) --ti;
    const int tj = ti + (t - (ti * NTILE - (ti * (ti - 1)) / 2)); // tj >= ti

    const int half = lane >> 4;            // 0: lanes 0-15, 1: lanes 16-31
    const int nl   = lane & 15;

    const float* __restrict__ xr = x + (size_t)n * D_DIM;        // SGPR base
    float* __restrict__ o        = out + (size_t)n * P_OUT;      // SGPR base

    // A: 16x4 f32. A.y = 0 kills K=1/K=3; lanes 16-31 put x into K=2, which is
    // harmless because B's K=2 row is forced to zero -> load A unconditionally.
    // B: 4x16 f32, only row K=0 (lanes 0-15, VGPR0) non-zero.
    const float av = xr[ti * 16 + nl];
    const float bv = xr[tj * 16 + nl];
    v2f A = {av, 0.0f};
    v2f B = {(half == 0) ? bv : 0.0f, 0.0f};
    v8f C = {0.0f, 0.0f, 0.0f, 0.0f, 0.0f, 0.0f, 0.0f, 0.0f};

    // D[m][n] = a_m * b_n in full f32 (matches reference math exactly)
    v8f Dm = __builtin_amdgcn_wmma_f32_16x16x4_f32(
        /*neg_a=*/false, A, /*neg_b=*/false, B,
        /*c_mod=*/(short)0, C, /*reuse_a=*/false, /*reuse_b=*/false);

    // D layout: VGPR v -> row m = v + 8*half, column = nl.
    // Packed-triangle offset: off(i) = i*D - i*(i-1)/2 + (j - i);
    // off(i+1) - off(i) = D - 1 - i  -> incremental 32-bit addressing.
    const int i0 = ti * 16 + (half << 3);
    const int j  = tj * 16 + nl;
    unsigned off = (unsigned)(i0 * D_DIM - (i0 * (i0 - 1)) / 2 + (j - i0));

    if (ti != tj) {
        // Off-diagonal tile (97% of tiles): every element valid, no masks.
        #pragma unroll
        for (int v = 0; v < 8; ++v) {
            __builtin_nontemporal_store(Dm[v], o + off);
            off += (unsigned)(D_DIM - 1 - (i0 + v));
        }
    } else {
        // Diagonal tile: keep only j >= i. (off wraps mod 2^32 when j < i,
        // but those lanes never store, and the wrap cancels in later adds.)
        #pragma unroll
        for (int v = 0; v < 8; ++v) {
            if (j >= i0 + v)
                __builtin_nontemporal_store(Dm[v], o + off);
            off += (unsigned)(D_DIM - 1 - (i0 + v));
        }
    }
}

extern "C" void kernel_launch(void* const* d_in, const int* in_sizes, int n_in,
                              void* d_out, int out_size, void* d_ws, size_t ws_size,
                              hipStream_t stream) {
    const float* x   = (const float*)d_in[0];
    float*       out = (float*)d_out;

    const int n_samples = in_sizes[0] / D_DIM;          // 64 for reference shapes
    const int blocks    = n_samples * BLK_PER_SAMPLE;   // 64 * 260 = 16640
    gram_triu_wmma<<<blocks, 256, 0, stream>>>(x, out);
}